// GraphEdgeRegressor_36197984370740
// MI455X (gfx1250) — compile-verified
//
#include <hip/hip_runtime.h>
#include <math.h>

// ---------------------------------------------------------------------------
// Problem constants (match reference)
// ---------------------------------------------------------------------------
#define NN   50000      // nodes
#define NE   800000     // edges (without self loops)
#define HID  64
#define EFD  192        // 3*HID
#define EH   128        // EDGE_HID

typedef __attribute__((ext_vector_type(2))) float v2f;
typedef __attribute__((ext_vector_type(8))) float v8f;

// ---------------------------------------------------------------------------
// float atomic-max via ordered-int trick (init must be -inf)
// ---------------------------------------------------------------------------
__device__ __forceinline__ void atomicMaxF32(float* addr, float val) {
    if (val >= 0.0f) {
        atomicMax((int*)addr, __float_as_int(val));
    } else {
        atomicMin((unsigned int*)addr, (unsigned int)__float_as_int(val));
    }
}

// ---------------------------------------------------------------------------
// Generic fp32 WMMA GEMM:  out[n x 64] = X[n x K] @ W[K x 64] (+ bias)
// block = 128 threads (4 waves), each block does a 16-row M tile,
// wave w covers output cols [16w, 16w+16).  Uses V_WMMA_F32_16X16X4_F32.
// ---------------------------------------------------------------------------
__global__ void gemm_n64_wmma(const float* __restrict__ X,
                              const float* __restrict__ W,
                              const float* __restrict__ bias,
                              float* __restrict__ out, int n, int K) {
    __shared__ float Xs[16 * 68];   // padded stride 68 -> conflict free A reads
    __shared__ float Ws[64 * 65];   // padded stride 65 -> conflict free B reads

    const int tid     = threadIdx.x;
    const int rowbase = blockIdx.x * 16;

    // cooperative loads
    for (int idx = tid; idx < K * 64; idx += 128) {
        Ws[(idx >> 6) * 65 + (idx & 63)] = W[idx];
    }
    for (int idx = tid; idx < 16 * K; idx += 128) {
        int r = idx / K, c = idx - r * K;
        int row = rowbase + r;
        Xs[r * 68 + c] = (row < n) ? X[row * K + c] : 0.0f;
    }
    __syncthreads();

    const int wave = tid >> 5;
    const int lane = tid & 31;
    const int half = lane >> 4;     // 0: K even pair, 1: K odd pair / rows +8
    const int ln   = lane & 15;
    const int col  = wave * 16 + ln;

    v8f acc;
    float b0 = bias ? bias[col] : 0.0f;
#pragma unroll
    for (int r = 0; r < 8; ++r) acc[r] = b0;

    for (int kb = 0; kb < K; kb += 4) {
        const int kk = kb + 2 * half;
        v2f a, b;
        a[0] = Xs[ln * 68 + kk];
        a[1] = Xs[ln * 68 + kk + 1];
        b[0] = Ws[kk * 65 + col];
        b[1] = Ws[(kk + 1) * 65 + col];
        acc = __builtin_amdgcn_wmma_f32_16x16x4_f32(false, a, false, b,
                                                    (short)0, acc, false, false);
    }

#pragma unroll
    for (int r = 0; r < 8; ++r) {
        int row = rowbase + r + 8 * half;
        if (row < n) out[row * 64 + col] = acc[r];
    }
}

// ---------------------------------------------------------------------------
// Per-layer init: agg = 0, m = -inf, denom = 0
// ---------------------------------------------------------------------------
__global__ void init_layer(float* __restrict__ agg, float* __restrict__ m,
                           float* __restrict__ denom, int n) {
    int t = blockIdx.x * blockDim.x + threadIdx.x;
    if (t < n * 64) agg[t] = 0.0f;
    if (t < n) { m[t] = -INFINITY; denom[t] = 0.0f; }
}

__global__ void zero_f32(float* __restrict__ p, int n) {
    int t = blockIdx.x * blockDim.x + threadIdx.x;
    if (t < n) p[t] = 0.0f;
}

// ---------------------------------------------------------------------------
// Pass 1: logits[e] = leaky_relu(xl[src]+xr[dst]) . att ; segment max into m
// edges E..E+N-1 are implicit self loops.
// ---------------------------------------------------------------------------
__global__ void edge_logits(const float* __restrict__ xl,
                            const float* __restrict__ xr,
                            const float* __restrict__ att,
                            const int* __restrict__ srcA,
                            const int* __restrict__ dstA,
                            float* __restrict__ m, float* __restrict__ elog,
                            int E, int N) {
    int e = blockIdx.x * blockDim.x + threadIdx.x;
    if (e >= E + N) return;
    int s = (e < E) ? srcA[e] : (e - E);
    int d = (e < E) ? dstA[e] : (e - E);
    const float4* xls = (const float4*)(xl + (size_t)s * 64);
    const float4* xrd = (const float4*)(xr + (size_t)d * 64);
    const float4* atv = (const float4*)att;
    float acc = 0.0f;
#pragma unroll
    for (int q = 0; q < 16; ++q) {
        float4 a = xls[q], r = xrd[q], w = atv[q];
        float t;
        t = a.x + r.x; t = (t > 0.0f) ? t : 0.2f * t; acc += t * w.x;
        t = a.y + r.y; t = (t > 0.0f) ? t : 0.2f * t; acc += t * w.y;
        t = a.z + r.z; t = (t > 0.0f) ? t : 0.2f * t; acc += t * w.z;
        t = a.w + r.w; t = (t > 0.0f) ? t : 0.2f * t; acc += t * w.w;
    }
    elog[e] = acc;
    atomicMaxF32(&m[d], acc);
}

// ---------------------------------------------------------------------------
// Pass 2: ex = exp(logit - m[dst]); denom[dst] += ex  (ex overwrites elog)
// ---------------------------------------------------------------------------
__global__ void edge_exp(float* __restrict__ elog, const float* __restrict__ m,
                         const int* __restrict__ dstA,
                         float* __restrict__ denom, int E, int N) {
    int e = blockIdx.x * blockDim.x + threadIdx.x;
    if (e >= E + N) return;
    int d = (e < E) ? dstA[e] : (e - E);
    float ex = expf(elog[e] - m[d]);
    elog[e] = ex;
    atomicAdd(&denom[d], ex);
}

// ---------------------------------------------------------------------------
// Pass 3: agg[dst] += (ex/denom[dst]) * xl[src]  (64 threads per edge)
// ---------------------------------------------------------------------------
__global__ void edge_scatter(const float* __restrict__ ex,
                             const float* __restrict__ denom,
                             const float* __restrict__ xl,
                             const int* __restrict__ srcA,
                             const int* __restrict__ dstA,
                             float* __restrict__ agg, int E, int N) {
    int t = blockIdx.x * blockDim.x + threadIdx.x;
    int e = t >> 6;
    int c = t & 63;
    if (e >= E + N) return;
    int s = (e < E) ? srcA[e] : (e - E);
    int d = (e < E) ? dstA[e] : (e - E);
    float alpha = ex[e] / denom[d];
    atomicAdd(&agg[(size_t)d * 64 + c], alpha * xl[(size_t)s * 64 + c]);
}

// ---------------------------------------------------------------------------
// h = relu(layernorm(agg + hres + bias) * g + b)  — one wave32 per node
// ---------------------------------------------------------------------------
__global__ void combine_ln(const float* __restrict__ agg,
                           const float* __restrict__ hres,
                           const float* __restrict__ bias,
                           const float* __restrict__ g,
                           const float* __restrict__ b,
                           float* __restrict__ h, int n) {
    int lane = threadIdx.x & 31;
    int node = blockIdx.x * 8 + (threadIdx.x >> 5);
    if (node >= n) return;                 // uniform per wave
    int c0 = lane, c1 = lane + 32;
    size_t base = (size_t)node * 64;
    float v0 = agg[base + c0] + hres[base + c0] + bias[c0];
    float v1 = agg[base + c1] + hres[base + c1] + bias[c1];
    float s = v0 + v1, s2 = v0 * v0 + v1 * v1;
#pragma unroll
    for (int off = 16; off > 0; off >>= 1) {
        s  += __shfl_xor(s,  off, 32);
        s2 += __shfl_xor(s2, off, 32);
    }
    float mu  = s * (1.0f / 64.0f);
    float var = s2 * (1.0f / 64.0f) - mu * mu;
    float inv = rsqrtf(var + 1e-5f);
    float o0 = (v0 - mu) * inv * g[c0] + b[c0];
    float o1 = (v1 - mu) * inv * g[c1] + b[c1];
    h[base + c0] = fmaxf(o0, 0.0f);
    h[base + c1] = fmaxf(o1, 0.0f);
}

// ---------------------------------------------------------------------------
// global_add_pool into gc[8][64]; hierarchical: LDS accumulators per block
// ---------------------------------------------------------------------------
#define POOL_NPB 512
__global__ void pool_kernel(const float* __restrict__ h,
                            const int* __restrict__ batch,
                            float* __restrict__ gc, int n) {
    __shared__ float loc[8 * 64];
    int tid = threadIdx.x;
    for (int i = tid; i < 512; i += 256) loc[i] = 0.0f;
    __syncthreads();
    int c  = tid & 63;
    int i0 = blockIdx.x * POOL_NPB + (tid >> 6);
    int iE = min(n, (int)(blockIdx.x + 1) * POOL_NPB);
    for (int i = i0; i < iE; i += 4) {
        atomicAdd(&loc[batch[i] * 64 + c], h[(size_t)i * 64 + c]);
    }
    __syncthreads();
    for (int i = tid; i < 512; i += 256) atomicAdd(&gc[i], loc[i]);
}

// ---------------------------------------------------------------------------
// Fused edge MLP with WMMA:
//   feat = [h[src] | h[dst] | gc[batch[src]]]  (48 edges x 192, gathered->LDS)
//   y1   = relu(feat @ W1 + b1)   (fp32 WMMA, 8 waves x 3 M-tiles, B reused)
//   out  = y1 @ W2 + b2
// y1 tile aliases the feat tile in LDS (feat dead after the k-loop).
// ---------------------------------------------------------------------------
#define EPB 48   // edges per block (3 WMMA M-tiles)
#define FS  196  // padded feat stride (192+4): bank-conflict-free A reads
#define YS  132  // padded y1 stride   (128+4)
__global__ void edge_mlp_wmma(const float* __restrict__ h,
                              const float* __restrict__ gc,
                              const int* __restrict__ batch,
                              const int* __restrict__ srcA,
                              const int* __restrict__ dstA,
                              const float* __restrict__ W1,
                              const float* __restrict__ b1,
                              const float* __restrict__ W2,
                              const float* __restrict__ b2,
                              float* __restrict__ out, int E) {
    __shared__ float smem[EPB * FS];          // feat, then (aliased) y1
    float* feat = smem;
    float* y1   = smem;                       // EPB*YS <= EPB*FS

    const int tid   = threadIdx.x;            // 256 threads = 8 waves
    const int ebase = blockIdx.x * EPB;

    // ---- gather phase (float4 vectorized: rows are 256B contiguous) ----
    for (int idx = tid; idx < EPB * (EFD / 4); idx += 256) {
        int er = idx / (EFD / 4);
        int q  = idx - er * (EFD / 4);        // float4 index within row
        int e  = ebase + er;
        float4 v = make_float4(0.0f, 0.0f, 0.0f, 0.0f);
        if (e < E) {
            int s = srcA[e];
            if (q < 16)      v = ((const float4*)(h + (size_t)s * 64))[q];
            else if (q < 32) v = ((const float4*)(h + (size_t)dstA[e] * 64))[q - 16];
            else             v = ((const float4*)(gc + batch[s] * 64))[q - 32];
        }
        *(float4*)(feat + er * FS + q * 4) = v;
    }
    __syncthreads();

    // ---- GEMM1: [48x192] @ [192x128], B fragment reused across 3 A tiles ----
    const int wave = tid >> 5;
    const int lane = tid & 31;
    const int half = lane >> 4;
    const int ln   = lane & 15;
    const int col  = wave * 16 + ln;          // 8 waves * 16 = 128 cols

    v8f acc0, acc1, acc2;
    float bb0 = b1[col];
#pragma unroll
    for (int r = 0; r < 8; ++r) { acc0[r] = bb0; acc1[r] = bb0; acc2[r] = bb0; }

#pragma unroll 4
    for (int kb = 0; kb < EFD; kb += 4) {
        const int kk = kb + 2 * half;
        v2f b, a0, a1, a2;
        b[0]  = W1[kk * EH + col];
        b[1]  = W1[(kk + 1) * EH + col];
        a0[0] = feat[ln * FS + kk];
        a0[1] = feat[ln * FS + kk + 1];
        a1[0] = feat[(ln + 16) * FS + kk];
        a1[1] = feat[(ln + 16) * FS + kk + 1];
        a2[0] = feat[(ln + 32) * FS + kk];
        a2[1] = feat[(ln + 32) * FS + kk + 1];
        acc0 = __builtin_amdgcn_wmma_f32_16x16x4_f32(false, a0, false, b,
                                                     (short)0, acc0, false, false);
        acc1 = __builtin_amdgcn_wmma_f32_16x16x4_f32(false, a1, false, b,
                                                     (short)0, acc1, false, false);
        acc2 = __builtin_amdgcn_wmma_f32_16x16x4_f32(false, a2, false, b,
                                                     (short)0, acc2, false, false);
    }
    __syncthreads();   // feat reads done; safe to overwrite with y1

#pragma unroll
    for (int r = 0; r < 8; ++r) {
        int row = r + 8 * half;
        y1[row * YS + col]        = fmaxf(acc0[r], 0.0f);
        y1[(row + 16) * YS + col] = fmaxf(acc1[r], 0.0f);
        y1[(row + 32) * YS + col] = fmaxf(acc2[r], 0.0f);
    }
    __syncthreads();

    // ---- GEMM2: [48x128] @ [128x1] ----
    if (tid < EPB) {
        int e = ebase + tid;
        if (e < E) {
            float s = b2[0];
#pragma unroll 8
            for (int k = 0; k < EH; ++k) s += y1[tid * YS + k] * W2[k];
            out[e] = s;
        }
    }
}

// ---------------------------------------------------------------------------
// Host orchestration
// ---------------------------------------------------------------------------
extern "C" void kernel_launch(void* const* d_in, const int* in_sizes, int n_in,
                              void* d_out, int out_size, void* d_ws, size_t ws_size,
                              hipStream_t stream) {
    (void)in_sizes; (void)n_in; (void)out_size; (void)ws_size;

    const float* x     = (const float*)d_in[0];
    const int*   ei    = (const int*)d_in[1];
    const int*   batch = (const int*)d_in[2];
    const int*   srcA  = ei;
    const int*   dstA  = ei + NE;

    // per-layer params: Wl, bl, Wr, br, att, res, bias, lng, lnb  (9 each, base 3)
    const float* Wl[3];  const float* bl[3];  const float* Wr[3]; const float* br[3];
    const float* att[3]; const float* res[3]; const float* bias[3];
    const float* lng[3]; const float* lnb[3];
    for (int l = 0; l < 3; ++l) {
        int o = 3 + 9 * l;
        Wl[l]   = (const float*)d_in[o + 0];
        bl[l]   = (const float*)d_in[o + 1];
        Wr[l]   = (const float*)d_in[o + 2];
        br[l]   = (const float*)d_in[o + 3];
        att[l]  = (const float*)d_in[o + 4];
        res[l]  = (const float*)d_in[o + 5];
        bias[l] = (const float*)d_in[o + 6];
        lng[l]  = (const float*)d_in[o + 7];
        lnb[l]  = (const float*)d_in[o + 8];
    }
    const float* W1 = (const float*)d_in[30];
    const float* b1 = (const float*)d_in[31];
    const float* W2 = (const float*)d_in[32];
    const float* b2 = (const float*)d_in[33];

    // workspace layout (fp32)
    const size_t N64 = (size_t)NN * 64;
    float* ws    = (float*)d_ws;
    float* h     = ws;               // N*64
    float* xl    = h    + N64;       // N*64
    float* xr    = xl   + N64;       // N*64
    float* hres  = xr   + N64;       // N*64
    float* agg   = hres + N64;       // N*64
    float* mmax  = agg  + N64;       // N
    float* denom = mmax + NN;        // N
    float* elog  = denom + NN;       // E+N
    float* gc    = elog + (NE + NN); // 8*64

    const int TOT = NE + NN;
    const dim3 b256(256), b128(128);
    const int gGemm    = (NN + 15) / 16;            // 3125
    const int gInit    = (NN * 64 + 255) / 256;
    const int gEdge    = (TOT + 255) / 256;
    const int gScatter = ((size_t)TOT * 64 + 255) / 256;
    const int gLN      = (NN + 7) / 8;
    const int gPool    = (NN + POOL_NPB - 1) / POOL_NPB;
    const int gMLP     = (NE + EPB - 1) / EPB;      // 16667

    for (int l = 0; l < 3; ++l) {
        const float* hin = (l == 0) ? x : h;
        const int K      = (l == 0) ? 32 : 64;
        gemm_n64_wmma<<<gGemm, b128, 0, stream>>>(hin, Wl[l], bl[l], xl, NN, K);
        gemm_n64_wmma<<<gGemm, b128, 0, stream>>>(hin, Wr[l], br[l], xr, NN, K);
        gemm_n64_wmma<<<gGemm, b128, 0, stream>>>(hin, res[l], nullptr, hres, NN, K);
        init_layer<<<gInit, b256, 0, stream>>>(agg, mmax, denom, NN);
        edge_logits<<<gEdge, b256, 0, stream>>>(xl, xr, att[l], srcA, dstA,
                                                mmax, elog, NE, NN);
        edge_exp<<<gEdge, b256, 0, stream>>>(elog, mmax, dstA, denom, NE, NN);
        edge_scatter<<<gScatter, b256, 0, stream>>>(elog, denom, xl, srcA, dstA,
                                                    agg, NE, NN);
        combine_ln<<<gLN, b256, 0, stream>>>(agg, hres, bias[l], lng[l], lnb[l],
                                             h, NN);
    }

    zero_f32<<<2, b256, 0, stream>>>(gc, 512);
    pool_kernel<<<gPool, b256, 0, stream>>>(h, batch, gc, NN);
    edge_mlp_wmma<<<gMLP, b256, 0, stream>>>(h, gc, batch, srcA, dstA,
                                             W1, b1, W2, b2, (float*)d_out, NE);
}